// EuclideanNet_35691178230066
// MI455X (gfx1250) — compile-verified
//
#include <hip/hip_runtime.h>
#include <hip/hip_bf16.h>
#include <math.h>

typedef __attribute__((ext_vector_type(2))) float v2f;
typedef __attribute__((ext_vector_type(8))) float v8f;

#define C_IN  23
#define NPTS  512
#define BATCH 16
#define HDIM  100
#define NTILE 7      // ceil(100/16)
#define TPAD  8      // NTILE padded to 8 for 32B per-lane loads

// Y00 / sqrt(N) = (1/(2*sqrt(pi))) / sqrt(512)
#define OUT_SCALE (0.28209479177387814f * 0.04419417382415922f)

__device__ __forceinline__ float relu1(float x) {
    // single v_med3_f32: clamp to [0, +inf) without canonicalize overhead
    return __builtin_amdgcn_fmed3f(x, 0.f, __builtin_inff());
}

// ---------------------------------------------------------------------------
// Phase 0: g2p[zb][col][t] = sum_j W2[h,j] * f[zb,j]  with h = t*16 + col,
// zero for h >= 100 and for the pad slot t == 7.  This transposed layout makes
// each lane's 7 h-tile values contiguous (32B) in phase 1.
// ---------------------------------------------------------------------------
__global__ void g2_kernel(const float* __restrict__ f,
                          const float* __restrict__ W2,
                          float* __restrict__ g2p) {
    int idx = blockIdx.x * blockDim.x + threadIdx.x;
    const int total = BATCH * NPTS * 16 * TPAD;
    if (idx >= total) return;
    int s   = idx & (16 * TPAD - 1);
    int zb  = idx >> 7;
    int col = s >> 3;
    int t   = s & 7;
    int h   = t * 16 + col;          // t==7 -> h >= 112 -> zero pad
    float sum = 0.f;
    if (h < HDIM) {
        const float* fr = f  + (size_t)zb * C_IN;
        const float* wr = W2 + (size_t)h  * C_IN;
        #pragma unroll
        for (int j = 0; j < C_IN; ++j) sum = fmaf(wr[j], fr[j], sum);
    }
    g2p[idx] = sum;
}

// ---------------------------------------------------------------------------
// Phase 1: the N^2 pair loop.  One block per (z, a-tile of 16); 8 waves each
// owning 64 b's.  Per b: per-lane distance + cosine basis -> WMMA A operand
// (16x4 f32), 7x v_wmma_f32_16x16x4_f32 against preloaded W1 tiles with b1
// folded into C, then med3-relu * g2 accumulated in VALU (pk_fma).
// ---------------------------------------------------------------------------
__global__ __launch_bounds__(256)
void pair_kernel(const float* __restrict__ geom,
                 const float* __restrict__ W1,
                 const float* __restrict__ b1,
                 const float* __restrict__ g2p,
                 float* __restrict__ convA) {
    const int atile = blockIdx.x;    // 0..31
    const int z     = blockIdx.y;    // 0..15
    const int tid   = threadIdx.x;
    const int wave  = tid >> 5;
    const int lane  = tid & 31;
    const int col   = lane & 15;     // A-matrix row / B,C,D column
    const int half  = lane >> 4;

    // geometry of this lane's 'a' row
    const int a = atile * 16 + col;
    const float gax = geom[(z * NPTS + a) * 3 + 0];
    const float gay = geom[(z * NPTS + a) * 3 + 1];
    const float gaz = geom[(z * NPTS + a) * 3 + 2];

    // Preload B tiles (W1: 4x16; lanes 0-15 hold K=0,1 in v0,v1, lanes 16-31
    // hold K=2,3) and C tiles (b1 bias broadcast down the 16 rows).
    v2f   Bt[NTILE];
    float Ct[NTILE];
    #pragma unroll
    for (int t = 0; t < NTILE; ++t) {
        int h = t * 16 + col;
        float w0 = 0.f, w1 = 0.f, c = 0.f;
        if (h < HDIM) {
            w0 = W1[(half ? 2 : 0) * HDIM + h];
            w1 = half ? 0.f : W1[1 * HDIM + h];
            c  = b1[h];
        }
        Bt[t].x = w0; Bt[t].y = w1; Ct[t] = c;
    }

    float acc[8];
    #pragma unroll
    for (int v = 0; v < 8; ++v) acc[v] = 0.f;

    __shared__ float sums[16];
    if (tid < 16) sums[tid] = 0.f;
    __syncthreads();

    const float kshift = half ? 2.f : 0.f;     // lane's K offsets: {0,1} or {2,3(pad)}
    const int   b0     = wave * (NPTS / 8);

    for (int bi = 0; bi < NPTS / 8; ++bi) {
        const int b = b0 + bi;
        const float* gb = geom + (size_t)(z * NPTS + b) * 3;
        float dx = gax - gb[0];
        float dy = gay - gb[1];
        float dz = gaz - gb[2];
        float rsq = fmaf(dx, dx, fmaf(dy, dy, dz * dz));
        float r   = sqrtf(fmaxf(rsq, 1e-12f));

        // basis_k(r) = |x|<1 ? cos(pi*x/2)^2 : 0,  x = r/1.5 - k
        //            = |x|<1 ? 0.5 + 0.5*cos(pi*x) : 0   (exact identity)
        float tt = r * (1.f / 1.5f);
        float x0 = tt - kshift;   // K = 0 (lanes 0-15) or K = 2 (lanes 16-31)
        float x1 = tt - 1.f;      // K = 1 (lanes 0-15); K = 3 is zero padding
        float c0 = 0.5f + 0.5f * __cosf(3.14159265358979f * x0);
        float c1 = 0.5f + 0.5f * __cosf(3.14159265358979f * x1);
        v2f A;
        A.x = (fabsf(x0) < 1.f) ? c0 : 0.f;
        A.y = (!half && (fabsf(x1) < 1.f)) ? c1 : 0.f;

        // one contiguous 32B per-lane load: this lane's g2 value for all 7
        // h-tiles (plus zero pad) -> 2x global_load_b128
        const v8f gvv = *(const v8f*)(g2p + ((size_t)(z * NPTS + b) * 16 + col) * TPAD);

        #pragma unroll
        for (int t = 0; t < NTILE; ++t) {
            float cv = Ct[t];
            v8f C = {cv, cv, cv, cv, cv, cv, cv, cv};
            // D = basis(16x4) * W1(4x16) + b1   — native f32 WMMA
            v8f D = __builtin_amdgcn_wmma_f32_16x16x4_f32(
                        false, A, false, Bt[t], (short)0, C, false, false);
            float gv = gvv[t];
            #pragma unroll
            for (int v = 0; v < 8; ++v)
                acc[v] = fmaf(relu1(D[v]), gv, acc[v]);
        }
    }

    // Reduce over the 16 h-columns (lanes) inside each half with xor shuffles.
    #pragma unroll
    for (int v = 0; v < 8; ++v) {
        float s = acc[v];
        s += __shfl_xor(s, 1, 32);
        s += __shfl_xor(s, 2, 32);
        s += __shfl_xor(s, 4, 32);
        s += __shfl_xor(s, 8, 32);
        acc[v] = s;
    }
    // Combine the 8 waves' partial sums per output row a = v + 8*half.
    if (col == 0) {
        #pragma unroll
        for (int v = 0; v < 8; ++v)
            atomicAdd(&sums[v + half * 8], acc[v]);
    }
    __syncthreads();
    if (tid < 16)
        convA[z * NPTS + atile * 16 + tid] = sums[tid] * OUT_SCALE;
}

// ---------------------------------------------------------------------------
// Phase 2: bias-dot correction (b2·f term) + tiny FC head. One block per z.
// ---------------------------------------------------------------------------
__global__ __launch_bounds__(128)
void head_kernel(const float* __restrict__ f,    const float* __restrict__ b2,
                 const float* __restrict__ convA,
                 const float* __restrict__ Wfc1, const float* __restrict__ bfc1,
                 const float* __restrict__ Wfc2, const float* __restrict__ bfc2,
                 const float* __restrict__ Wfc3, const float* __restrict__ bfc3,
                 float* __restrict__ out) {
    const int z   = blockIdx.x;
    const int tid = threadIdx.x;
    __shared__ float red[128];
    __shared__ float convs[NPTS];
    __shared__ float h1s[30];
    __shared__ float h2s[10];

    // sfb2 = sum_b f[z,b,:]·b2   (same additive term for every a)
    float p = 0.f;
    for (int idx = tid; idx < NPTS * C_IN; idx += 128) {
        int j = idx % C_IN;
        p = fmaf(f[(size_t)z * NPTS * C_IN + idx], b2[j], p);
    }
    red[tid] = p;
    __syncthreads();
    for (int s = 64; s > 0; s >>= 1) {
        if (tid < s) red[tid] += red[tid + s];
        __syncthreads();
    }
    const float c = red[0] * OUT_SCALE;

    for (int aIdx = tid; aIdx < NPTS; aIdx += 128)
        convs[aIdx] = convA[z * NPTS + aIdx] + c;
    __syncthreads();

    if (tid < 30) {
        float s = bfc1[tid];
        for (int aa = 0; aa < NPTS; ++aa)
            s = fmaf(convs[aa], Wfc1[aa * 30 + tid], s);
        h1s[tid] = relu1(s);
    }
    __syncthreads();
    if (tid < 10) {
        float s = bfc2[tid];
        #pragma unroll
        for (int u = 0; u < 30; ++u) s = fmaf(h1s[u], Wfc2[u * 10 + tid], s);
        h2s[tid] = relu1(s);
    }
    __syncthreads();
    if (tid == 0) {
        float s = bfc3[0];
        #pragma unroll
        for (int u = 0; u < 10; ++u) s = fmaf(h2s[u], Wfc3[u], s);
        out[z] = relu1(s);
    }
}

// ---------------------------------------------------------------------------
extern "C" void kernel_launch(void* const* d_in, const int* in_sizes, int n_in,
                              void* d_out, int out_size, void* d_ws, size_t ws_size,
                              hipStream_t stream) {
    const float* f    = (const float*)d_in[0];
    const float* geom = (const float*)d_in[1];
    const float* W1   = (const float*)d_in[2];
    const float* b1   = (const float*)d_in[3];
    const float* W2   = (const float*)d_in[4];
    const float* b2   = (const float*)d_in[5];
    const float* Wfc1 = (const float*)d_in[6];
    const float* bfc1 = (const float*)d_in[7];
    const float* Wfc2 = (const float*)d_in[8];
    const float* bfc2 = (const float*)d_in[9];
    const float* Wfc3 = (const float*)d_in[10];
    const float* bfc3 = (const float*)d_in[11];
    float* out = (float*)d_out;

    float* g2p   = (float*)d_ws;                                // 16*512*16*8 f32 = 4.19 MB
    float* convA = g2p + (size_t)BATCH * NPTS * 16 * TPAD;      // 16*512 f32

    g2_kernel<<<(BATCH * NPTS * 16 * TPAD + 255) / 256, 256, 0, stream>>>(f, W2, g2p);
    pair_kernel<<<dim3(NPTS / 16, BATCH), 256, 0, stream>>>(geom, W1, b1, g2p, convA);
    head_kernel<<<BATCH, 128, 0, stream>>>(f, b2, convA,
                                           Wfc1, bfc1, Wfc2, bfc2, Wfc3, bfc3, out);
}